// FinalSphericalChebBN_78572131713679
// MI455X (gfx1250) — compile-verified
//
#include <hip/hip_runtime.h>

typedef float v2f __attribute__((ext_vector_type(2)));
typedef float v8f __attribute__((ext_vector_type(8)));

constexpr int V    = 12288;
constexpr int C    = 128;    // C_IN == C_OUT
constexpr int KCH  = 4;      // Chebyshev degree
constexpr int DEG  = 9;
constexpr int B    = 8;
constexpr int M    = B * V;        // 98304 rows
constexpr int KRED = C * KCH;      // 512 reduction
constexpr float BN_EPS = 1e-5f;

// workspace layout (in floats)
constexpr size_t WP_OFF   = 0;                          // permuted weight: 512*128
constexpr size_t GSUM_OFF = WP_OFF + (size_t)KRED * C;  // 128 sums
constexpr size_t GSQ_OFF  = GSUM_OFF + 128;             // 128 sumsq
constexpr size_t T1_OFF   = GSQ_OFF + 128;
constexpr size_t TSZ      = (size_t)M * C;              // 12,582,912 floats each
constexpr size_t T2_OFF   = T1_OFF + TSZ;
constexpr size_t T3_OFF   = T2_OFF + TSZ;
constexpr size_t CONV_OFF = T3_OFF + TSZ;

// ---------------------------------------------------------------------------
// Prep: permute weight[(c*4+k)*128+o] -> Wp with K = k*128+c, (K,K+1) column-
// pair interleaved: Wp[(K>>1)*256 + o*2 + (K&1)]. Also zero the BN stats.
// ---------------------------------------------------------------------------
__global__ __launch_bounds__(256) void prep_kernel(const float* __restrict__ w,
                                                   float* __restrict__ wp,
                                                   float* __restrict__ stats) {
  int gid  = blockIdx.x * 256 + threadIdx.x;   // 0..65535
  int Kidx = gid >> 7;                          // 0..511
  int n    = gid & 127;
  int k    = Kidx >> 7;                         // 0..3
  int c    = Kidx & 127;
  wp[((size_t)(Kidx >> 1)) * 256 + (n << 1) + (Kidx & 1)] =
      w[((c << 2) + k) * 128 + n];
  if (gid < 256) stats[gid] = 0.0f;             // gsum(128) + gsq(128), contiguous
}

// ---------------------------------------------------------------------------
// Chebyshev SPMM: out[b,v,:] = scale * sum_j vals[v*9+j] * prev[b,col_j,:]
//                              + add * prev2[b,v,:]
// One wave per (b,v); lane covers 4 channels via float4. Cols/vals are wave-
// uniform (indexed by blockIdx.x) -> scalar loads; gathers are 512B coalesced.
// ---------------------------------------------------------------------------
__global__ __launch_bounds__(256) void cheb_spmm_kernel(
    const float* __restrict__ prev, const float* __restrict__ prev2,
    const int* __restrict__ cols, const float* __restrict__ vals,
    float* __restrict__ out, float scale, float add) {
  int v    = blockIdx.x;
  int b    = threadIdx.y;
  int lane = threadIdx.x;
  size_t rowbase = ((size_t)b * V + v) * C + lane * 4;
  float4 acc = make_float4(0.f, 0.f, 0.f, 0.f);
#pragma unroll
  for (int j = 0; j < DEG; ++j) {
    int   col = cols[v * DEG + j];
    float wv  = vals[v * DEG + j];
    const float4 y = *(const float4*)(prev + ((size_t)b * V + col) * C + lane * 4);
    acc.x += wv * y.x; acc.y += wv * y.y; acc.z += wv * y.z; acc.w += wv * y.w;
  }
  float4 r;
  r.x = scale * acc.x; r.y = scale * acc.y;
  r.z = scale * acc.z; r.w = scale * acc.w;
  if (add != 0.0f) {
    const float4 p = *(const float4*)(prev2 + rowbase);
    r.x += add * p.x; r.y += add * p.y; r.z += add * p.z; r.w += add * p.w;
  }
  *(float4*)(out + rowbase) = r;
}

// ---------------------------------------------------------------------------
// GEMM: conv[m, o] = sum_k sum_c T_k[m, c] * W[(c*4+k), o]  via f32 WMMA.
// Block = 8 waves, one 16-row M tile staged in LDS (row stride 516 dwords ->
// conflict-free b64 A-fragment reads). Wave w owns N columns [16w, 16w+16).
// 128 x v_wmma_f32_16x16x4_f32 per wave. Epilogue: store conv + BN stats.
// ---------------------------------------------------------------------------
constexpr int LS = 516;  // LDS row stride (dwords): LS % 64 == 4 -> 64 distinct banks

__global__ __launch_bounds__(256) void gemm_wmma_kernel(
    const float* __restrict__ T0, const float* __restrict__ T1,
    const float* __restrict__ T2, const float* __restrict__ T3,
    const float* __restrict__ wp, float* __restrict__ conv,
    float* __restrict__ gsum, float* __restrict__ gsq) {
  __shared__ float Alds[16 * LS];
  __shared__ float bs[128];
  __shared__ float bq[128];

  const int tid  = threadIdx.x;
  const int wave = tid >> 5;
  const int lane = tid & 31;
  const int r    = lane & 15;   // M row (A) / N col (B,C)
  const int g    = lane >> 4;   // lane-group: K sub-offset
  const int m0   = blockIdx.x * 16;

  // cooperative A-tile load: 16 rows x 512 K (K = k*128 + c), float4 per thread x8
#pragma unroll
  for (int i = 0; i < 8; ++i) {
    int idx = tid + 256 * i;          // 0..2047 float4s
    int row = idx >> 7;               // 0..15
    int q   = idx & 127;
    int k   = q >> 5;                 // source term 0..3
    int c4  = (q & 31) << 2;          // channel base
    const float* src = (k == 0) ? T0 : (k == 1) ? T1 : (k == 2) ? T2 : T3;
    float4 val = *(const float4*)(src + ((size_t)(m0 + row)) * C + c4);
    *(float4*)(&Alds[row * LS + (k << 7) + c4]) = val;
  }
  if (tid < 128) { bs[tid] = 0.0f; bq[tid] = 0.0f; }
  __syncthreads();

  const int n0 = wave << 4;
  v8f acc = {0.f, 0.f, 0.f, 0.f, 0.f, 0.f, 0.f, 0.f};
  const float* arow  = &Alds[r * LS + 2 * g];
  const float* bbase = wp + g * 256 + ((n0 + r) << 1);

#pragma unroll 4
  for (int kk = 0; kk < KRED; kk += 4) {
    v2f a  = *(const v2f*)(arow + kk);            // ds_load_b64, conflict-free
    v2f bf = *(const v2f*)(bbase + (kk << 7));    // global b64, coalesced, L2-hot
    acc = __builtin_amdgcn_wmma_f32_16x16x4_f32(
        /*neg_a=*/false, a, /*neg_b=*/false, bf,
        /*c_mod=*/(short)0, acc, /*reuse_a=*/false, /*reuse_b=*/false);
  }

  // epilogue: write conv tile, accumulate per-channel BN partial stats
  float s = 0.0f, s2 = 0.0f;
#pragma unroll
  for (int j = 0; j < 8; ++j) {
    int   Mrow = j + (g << 3);                    // C/D layout: VGPR j -> M = j + 8*g
    float val  = acc[j];
    conv[((size_t)(m0 + Mrow)) * C + n0 + r] = val;
    s += val; s2 += val * val;
  }
  atomicAdd(&bs[n0 + r], s);
  atomicAdd(&bq[n0 + r], s2);
  __syncthreads();
  if (tid < 128) {
    atomicAdd(&gsum[tid], bs[tid]);
    atomicAdd(&gsq[tid],  bq[tid]);
  }
}

// ---------------------------------------------------------------------------
// BN + transpose: out[b,o,v] = (conv[b,v,o]-mean[o])*rsqrt(var[o]+eps)*gamma+beta
// 32x128 tile via LDS (stride 129 -> conflict-free transpose reads); writes are
// 32-element contiguous runs along v.
// ---------------------------------------------------------------------------
__global__ __launch_bounds__(256) void bn_out_kernel(
    const float* __restrict__ conv, const float* __restrict__ gsum,
    const float* __restrict__ gsq, const float* __restrict__ gamma,
    const float* __restrict__ beta, float* __restrict__ out) {
  __shared__ float t[32 * 129];
  __shared__ float sc[128];
  __shared__ float sh[128];
  const int tid = threadIdx.x;
  const int blk = blockIdx.x;
  const int b   = blk / (V / 32);
  const int v0  = (blk % (V / 32)) * 32;
  constexpr float invN = 1.0f / (float)M;

  if (tid < 128) {
    float mean = gsum[tid] * invN;
    float var  = gsq[tid] * invN - mean * mean;   // biased, matches jnp.var
    float rs   = rsqrtf(var + BN_EPS);
    float scl  = gamma[tid] * rs;
    sc[tid] = scl;
    sh[tid] = beta[tid] - mean * scl;
  }
#pragma unroll
  for (int i = 0; i < 16; ++i) {
    int idx = tid + 256 * i;          // 4096 elements
    int row = idx >> 7;
    int c   = idx & 127;
    t[row * 129 + c] = conv[((size_t)b * V + v0 + row) * C + c];
  }
  __syncthreads();
#pragma unroll
  for (int i = 0; i < 16; ++i) {
    int idx = tid + 256 * i;
    int o   = idx >> 5;
    int vv  = idx & 31;
    out[(size_t)b * C * V + (size_t)o * V + v0 + vv] = t[vv * 129 + o] * sc[o] + sh[o];
  }
}

// ---------------------------------------------------------------------------
extern "C" void kernel_launch(void* const* d_in, const int* in_sizes, int n_in,
                              void* d_out, int out_size, void* d_ws, size_t ws_size,
                              hipStream_t stream) {
  const float* x        = (const float*)d_in[0];
  // d_in[1] = lap_rows: structurally repeat(arange(V), 9) -> implicit, unused
  const int*   lap_cols = (const int*)  d_in[2];
  const float* lap_vals = (const float*)d_in[3];
  const float* weight   = (const float*)d_in[4];
  const float* gamma    = (const float*)d_in[5];
  const float* beta     = (const float*)d_in[6];

  float* ws   = (float*)d_ws;
  float* wp   = ws + WP_OFF;
  float* gsum = ws + GSUM_OFF;
  float* gsq  = ws + GSQ_OFF;
  float* t1   = ws + T1_OFF;
  float* t2   = ws + T2_OFF;
  float* t3   = ws + T3_OFF;
  float* conv = ws + CONV_OFF;
  float* out  = (float*)d_out;

  // 1. permute weight + zero BN stats (re-run every call: graph-replay safe)
  prep_kernel<<<(KRED * C) / 256, 256, 0, stream>>>(weight, wp, gsum);

  // 2. Chebyshev recursion: T1 = Lx; T2 = 2*L*T1 - x; T3 = 2*L*T2 - T1
  dim3 cblk(32, 8);
  cheb_spmm_kernel<<<V, cblk, 0, stream>>>(x,  x,  lap_cols, lap_vals, t1, 1.0f,  0.0f);
  cheb_spmm_kernel<<<V, cblk, 0, stream>>>(t1, x,  lap_cols, lap_vals, t2, 2.0f, -1.0f);
  cheb_spmm_kernel<<<V, cblk, 0, stream>>>(t2, t1, lap_cols, lap_vals, t3, 2.0f, -1.0f);

  // 3. fp32 WMMA GEMM + BN stat accumulation
  gemm_wmma_kernel<<<M / 16, 256, 0, stream>>>(x, t1, t2, t3, wp, conv, gsum, gsq);

  // 4. BN finalize + [B,V,C] -> [B,C,V] transpose
  bn_out_kernel<<<B * (V / 32), 256, 0, stream>>>(conv, gsum, gsq, gamma, beta, out);
}